// two_layer_GCN_32985348833474
// MI455X (gfx1250) — compile-verified
//
#include <hip/hip_runtime.h>
#include <hip/hip_bf16.h>

typedef __attribute__((ext_vector_type(2))) float v2f;
typedef __attribute__((ext_vector_type(8))) float v8f;

#define K1 512
#define N1 128
#define K2 128
#define N2 40

// ---------------------------------------------------------------------------
// GEMM1: h1[M,128] = feature[M,512] @ W1[512,128], f32 WMMA 16x16x4.
// Block = 256 threads (8 waves), 16 rows per block. A tile staged in LDS with
// padded stride 516 dwords (516 % 64 = 4 -> conflict-free banks; compiler
// merges the pair loads into ds_load_2addr_b64). Two interleaved accumulators
// break the WMMA->WMMA RAW chain on SRC2.
// ---------------------------------------------------------------------------
__global__ void __launch_bounds__(256)
gcn_gemm1_wmma(const float* __restrict__ A, const float* __restrict__ B,
               float* __restrict__ C, int M) {
    constexpr int LDA = K1 + 4;          // 516 dwords
    __shared__ float sA[16 * LDA];       // 33 KB

    const int tid  = threadIdx.x;
    const int row0 = blockIdx.x * 16;

    for (int i = tid; i < 16 * K1; i += 256) {
        const int r = i >> 9;
        const int c = i & (K1 - 1);
        const int gr = row0 + r;
        sA[r * LDA + c] = (gr < M) ? A[(size_t)gr * K1 + c] : 0.0f;
    }
    __syncthreads();

    const int wave = tid >> 5;           // 0..7 -> column tile
    const int lane = tid & 31;
    const int hf   = lane >> 4;
    const int mn   = lane & 15;
    const int tn0  = wave * 16;

    v8f acc0 = {}, acc1 = {};
    #pragma unroll 4
    for (int k0 = 0; k0 < K1; k0 += 8) {
        const int ka = k0 + 2 * hf;
        v2f a0, b0, a1, b1;
        a0.x = sA[mn * LDA + ka];
        a0.y = sA[mn * LDA + ka + 1];
        a1.x = sA[mn * LDA + ka + 4];
        a1.y = sA[mn * LDA + ka + 5];
        b0.x = B[(size_t)ka * N1 + tn0 + mn];
        b0.y = B[(size_t)(ka + 1) * N1 + tn0 + mn];
        b1.x = B[(size_t)(ka + 4) * N1 + tn0 + mn];
        b1.y = B[(size_t)(ka + 5) * N1 + tn0 + mn];
        acc0 = __builtin_amdgcn_wmma_f32_16x16x4_f32(
            false, a0, false, b0, (short)0, acc0, false, false);
        acc1 = __builtin_amdgcn_wmma_f32_16x16x4_f32(
            false, a1, false, b1, (short)0, acc1, false, false);
    }
    const v8f acc = acc0 + acc1;

    float* __restrict__ cp = C + (size_t)(row0 + 8 * hf) * N1 + tn0 + mn;
    if (row0 + 16 <= M) {                // uniform fast path (M % 16 == 0 case)
        #pragma unroll
        for (int r = 0; r < 8; ++r) cp[(size_t)r * N1] = acc[r];
    } else {
        #pragma unroll
        for (int r = 0; r < 8; ++r)
            if (row0 + 8 * hf + r < M) cp[(size_t)r * N1] = acc[r];
    }
}

// ---------------------------------------------------------------------------
// GEMM2: h2[M,40] = relu(agg1[M,128]) @ W2[128,40]. ReLU fused into staging.
// Block = 96 threads (3 waves); column tiles cover 0..47, guarded at n>=40.
// ---------------------------------------------------------------------------
__global__ void __launch_bounds__(96)
gcn_gemm2_wmma(const float* __restrict__ A, const float* __restrict__ B,
               float* __restrict__ C, int M) {
    constexpr int LDA = K2 + 4;          // 132 dwords (132 % 64 = 4)
    __shared__ float sA[16 * LDA];

    const int tid  = threadIdx.x;
    const int row0 = blockIdx.x * 16;

    for (int i = tid; i < 16 * K2; i += 96) {
        const int r = i >> 7;
        const int c = i & (K2 - 1);
        const int gr = row0 + r;
        float v = (gr < M) ? A[(size_t)gr * K2 + c] : 0.0f;
        sA[r * LDA + c] = v > 0.0f ? v : 0.0f;   // fused ReLU (layer-1 act)
    }
    __syncthreads();

    const int wave = tid >> 5;           // 0..2
    const int lane = tid & 31;
    const int hf   = lane >> 4;
    const int mn   = lane & 15;
    const int tn0  = wave * 16;
    const int n    = tn0 + mn;
    const bool nok = (n < N2);
    const int nc   = nok ? n : (N2 - 1); // clamp: avoid OOB, value masked below
    const float bm = nok ? 1.0f : 0.0f;

    v8f acc0 = {}, acc1 = {};
    #pragma unroll 4
    for (int k0 = 0; k0 < K2; k0 += 8) {
        const int ka = k0 + 2 * hf;
        v2f a0, b0, a1, b1;
        a0.x = sA[mn * LDA + ka];
        a0.y = sA[mn * LDA + ka + 1];
        a1.x = sA[mn * LDA + ka + 4];
        a1.y = sA[mn * LDA + ka + 5];
        b0.x = bm * B[(size_t)ka * N2 + nc];
        b0.y = bm * B[(size_t)(ka + 1) * N2 + nc];
        b1.x = bm * B[(size_t)(ka + 4) * N2 + nc];
        b1.y = bm * B[(size_t)(ka + 5) * N2 + nc];
        acc0 = __builtin_amdgcn_wmma_f32_16x16x4_f32(
            false, a0, false, b0, (short)0, acc0, false, false);
        acc1 = __builtin_amdgcn_wmma_f32_16x16x4_f32(
            false, a1, false, b1, (short)0, acc1, false, false);
    }
    const v8f acc = acc0 + acc1;

    float* __restrict__ cp = C + (size_t)(row0 + 8 * hf) * N2 + n;
    if (row0 + 16 <= M) {                // uniform row fast path
        if (nok) {
            #pragma unroll
            for (int r = 0; r < 8; ++r) cp[(size_t)r * N2] = acc[r];
        }
    } else {
        #pragma unroll
        for (int r = 0; r < 8; ++r)
            if (nok && (row0 + 8 * hf + r < M)) cp[(size_t)r * N2] = acc[r];
    }
}

// ---------------------------------------------------------------------------
// Edge scatter, layer 1: agg[dst[e], :128] += h[src[e], :128] * w[e].
// One wave per edge; lane handles 4 contiguous floats (float4 gather,
// 4x global_atomic_add_f32, non-returning -> STOREcnt only, no result wait).
// h1/agg1 (51 MB each) are L2-resident on the 192 MB L2.
// ---------------------------------------------------------------------------
__global__ void __launch_bounds__(256)
gcn_scatter128(const float* __restrict__ h, const int* __restrict__ src,
               const int* __restrict__ dst, const float* __restrict__ w,
               float* __restrict__ agg, int E) {
    const int e = blockIdx.x * 8 + (threadIdx.x >> 5);
    if (e >= E) return;
    const int lane = threadIdx.x & 31;
    const int s = src[e];
    const int d = dst[e];
    const float we = w[e];
    const float4 v = *(const float4*)(h + (size_t)s * N1 + lane * 4);
    float* o = agg + (size_t)d * N1 + lane * 4;
    atomicAdd(o + 0, v.x * we);
    atomicAdd(o + 1, v.y * we);
    atomicAdd(o + 2, v.z * we);
    atomicAdd(o + 3, v.w * we);
}

// Edge scatter, layer 2: out[dst[e], :40] += h2[src[e], :40] * w[e].
__global__ void __launch_bounds__(256)
gcn_scatter40(const float* __restrict__ h, const int* __restrict__ src,
              const int* __restrict__ dst, const float* __restrict__ w,
              float* __restrict__ out, int E) {
    const int e = blockIdx.x * 8 + (threadIdx.x >> 5);
    if (e >= E) return;
    const int lane = threadIdx.x & 31;
    const int s = src[e];
    const int d = dst[e];
    const float we = w[e];
    #pragma unroll
    for (int c = lane; c < N2; c += 32)
        atomicAdd(out + (size_t)d * N2 + c, h[(size_t)s * N2 + c] * we);
}

__global__ void gcn_zero_f4(float4* __restrict__ p, long n4) {
    const long i = (long)blockIdx.x * blockDim.x + threadIdx.x;
    if (i < n4) p[i] = make_float4(0.f, 0.f, 0.f, 0.f);
}

extern "C" void kernel_launch(void* const* d_in, const int* in_sizes, int n_in,
                              void* d_out, int out_size, void* d_ws, size_t ws_size,
                              hipStream_t stream) {
    const float* feature = (const float*)d_in[0];   // [N,512]
    const int*   src     = (const int*)  d_in[1];   // [E]
    const int*   dst     = (const int*)  d_in[2];   // [E]
    const float* ew      = (const float*)d_in[3];   // [E]
    const float* W1      = (const float*)d_in[4];   // [512,128]
    const float* W2      = (const float*)d_in[5];   // [128,40]
    float*       out     = (float*)d_out;           // [N,40]

    const int N = in_sizes[0] / K1;                 // 100000
    const int E = in_sizes[1];                      // 1600000

    // workspace layout: h1 [N*128] | agg1 [N*128];  h2 [N*40] aliases h1
    float* h1   = (float*)d_ws;
    float* agg1 = h1 + (size_t)N * N1;
    float* h2   = h1;

    const int rowBlocks  = (N + 15) / 16;
    const int edgeBlocks = (E + 7) / 8;

    // layer 1
    gcn_gemm1_wmma<<<rowBlocks, 256, 0, stream>>>(feature, W1, h1, N);
    {
        long n4 = ((long)N * N1) / 4;
        gcn_zero_f4<<<(int)((n4 + 255) / 256), 256, 0, stream>>>((float4*)agg1, n4);
    }
    gcn_scatter128<<<edgeBlocks, 256, 0, stream>>>(h1, src, dst, ew, agg1, E);

    // layer 2 (ReLU fused into gemm2's A staging); h2 overwrites h1 (done with it)
    gcn_gemm2_wmma<<<rowBlocks, 96, 0, stream>>>(agg1, W2, h2, N);
    {
        long n4 = ((long)N * N2) / 4;
        gcn_zero_f4<<<(int)((n4 + 255) / 256), 256, 0, stream>>>((float4*)out, n4);
    }
    gcn_scatter40<<<edgeBlocks, 256, 0, stream>>>(h2, src, dst, ew, out, E);
}